// GlobalSemanticsAggregator_66486093742336
// MI455X (gfx1250) — compile-verified
//
#include <hip/hip_runtime.h>
#include <math.h>

// Problem constants (from reference setup_inputs)
#define TT 512
#define BB 64
#define DD 256
#define FF 256
#define ALPHA 0.2f

typedef float v2f __attribute__((ext_vector_type(2)));
typedef float v8f __attribute__((ext_vector_type(8)));

static __device__ __forceinline__ float lrelu(float x) {
    return x > 0.0f ? x : ALPHA * x;
}

// ---------------------------------------------------------------------------
// Kernel A: wa1[d] = sum_f W[d,f]*a1[f] ; wa2[d] = sum_f W[d,f]*a2[f]
// W is (D,F) row-major. One block, 256 threads (thread = d).
// ---------------------------------------------------------------------------
__global__ void k_wa(const float* __restrict__ W,
                     const float* __restrict__ a1,
                     const float* __restrict__ a2,
                     float* __restrict__ wa /* [2*DD]: wa1 | wa2 */) {
    __shared__ float sa1[FF], sa2[FF];
    int tid = threadIdx.x;
    sa1[tid] = a1[tid];
    sa2[tid] = a2[tid];
    __syncthreads();
    const float* row = W + (size_t)tid * FF;
    float s1 = 0.f, s2 = 0.f;
    #pragma unroll 8
    for (int f = 0; f < FF; ++f) {
        float wv = row[f];
        s1 += wv * sa1[f];
        s2 += wv * sa2[f];
    }
    wa[tid]      = s1;
    wa[DD + tid] = s2;
}

// ---------------------------------------------------------------------------
// Kernel B: p1[b*T+t] = x[t,b,:] . wa1 ; p2 likewise.
// x layout (T,B,D). One wave32 per row; 8 waves per block (256 thr).
// ---------------------------------------------------------------------------
__global__ void k_rowdots(const float* __restrict__ x,
                          const float* __restrict__ wa,
                          float* __restrict__ p1,
                          float* __restrict__ p2) {
    int wave = threadIdx.x >> 5;
    int lane = threadIdx.x & 31;
    int r = blockIdx.x * 8 + wave;          // r = t*B + b, r < T*B
    const float* row = x + (size_t)r * DD;
    float s1 = 0.f, s2 = 0.f;
    #pragma unroll
    for (int k = lane; k < DD; k += 32) {
        float xv = row[k];
        s1 += xv * wa[k];
        s2 += xv * wa[DD + k];
    }
    #pragma unroll
    for (int off = 16; off > 0; off >>= 1) {
        s1 += __shfl_xor(s1, off, 32);
        s2 += __shfl_xor(s2, off, 32);
    }
    if (lane == 0) {
        int t = r >> 6;          // r / B
        int b = r & (BB - 1);    // r % B
        p1[b * TT + t] = s1;
        p2[b * TT + t] = s2;
    }
}

// ---------------------------------------------------------------------------
// Kernel C: per-batch attention coefficient vector Dcoef[b,t].
// For each window w: cnt = max(turns-w+2,0); f1/f2 = window means of p1/p2;
// c_j = (1/cnt) * sum_{i<cnt} softmax_j(lrelu(f1_i+f2_j));
// coef[t] += (1/w) * sum_{k<w} c_{t-k}.  Final: Dcoef = coef / vws.
// One block (256 threads) per b.
// ---------------------------------------------------------------------------
__global__ void k_attn(const float* __restrict__ p1,
                       const float* __restrict__ p2,
                       const int* __restrict__ turns,
                       float* __restrict__ Dcoef) {
    __shared__ float sp1[TT], sp2[TT];
    __shared__ float f1a[TT], f2a[TT];
    __shared__ float rJ[TT];      // 1/(Z_i * denom)
    __shared__ float cjA[TT];
    __shared__ float coef[TT];
    __shared__ float red[256];

    int b = blockIdx.x;
    int tid = threadIdx.x;
    for (int t = tid; t < TT; t += 256) {
        sp1[t] = p1[b * TT + t];
        sp2[t] = p2[b * TT + t];
        coef[t] = 0.f;
    }
    __syncthreads();

    int turn = turns[b];
    float vws = 0.f;

    for (int w = 1; w <= 3; ++w) {
        int cnt = turn - w + 2;
        if (cnt < 0) cnt = 0;
        if (cnt == 0) continue;            // uniform across block
        vws += 1.f;
        float invw = 1.f / (float)w;
        float denom = (float)cnt;

        // windowed means of p1/p2 (valid rows never touch padding: i+w-1<=turn<T)
        for (int i = tid; i < cnt; i += 256) {
            float s1 = 0.f, s2 = 0.f;
            for (int k = 0; k < w; ++k) { s1 += sp1[i + k]; s2 += sp2[i + k]; }
            f1a[i] = s1 * invw;
            f2a[i] = s2 * invw;
        }
        __syncthreads();

        // max over valid f2 (lrelu monotone => rowmax_i = lrelu(f1_i + max f2))
        float lm = -3.4e38f;
        for (int j = tid; j < cnt; j += 256) lm = fmaxf(lm, f2a[j]);
        red[tid] = lm;
        __syncthreads();
        for (int s = 128; s > 0; s >>= 1) {
            if (tid < s) red[tid] = fmaxf(red[tid], red[tid + s]);
            __syncthreads();
        }
        float mx = red[0];

        // Pass 1: Z_i and fold 1/denom
        for (int i = tid; i < cnt; i += 256) {
            float f1 = f1a[i];
            float rm = lrelu(f1 + mx);
            float z = 0.f;
            for (int j = 0; j < cnt; ++j) {
                float e = lrelu(f1 + f2a[j]);
                z += __expf(e - rm);
            }
            rJ[i] = 1.f / (z * denom);
        }
        __syncthreads();

        // Pass 2: c_j = sum_i exp(e_ij - rm_i) * rJ[i]
        for (int j = tid; j < cnt; j += 256) {
            float f2 = f2a[j];
            float acc = 0.f;
            for (int i = 0; i < cnt; ++i) {
                float f1 = f1a[i];
                float rm = lrelu(f1 + mx);
                float e  = lrelu(f1 + f2);
                acc += __expf(e - rm) * rJ[i];
            }
            cjA[j] = acc;
        }
        __syncthreads();

        // Spread window-mean weights back onto raw rows
        for (int t = tid; t < TT; t += 256) {
            float add = 0.f;
            for (int k = 0; k < w; ++k) {
                int j = t - k;
                if (j >= 0 && j < cnt) add += cjA[j];
            }
            coef[t] += add * invw;
        }
        __syncthreads();
    }

    float ivws = (vws > 0.f) ? (1.f / vws) : 0.f;
    for (int t = tid; t < TT; t += 256)
        Dcoef[b * TT + t] = coef[t] * ivws;
}

// ---------------------------------------------------------------------------
// Kernel D: y[b,d] = sum_t Dcoef[b,t] * x[t,b,d].  Block per b, thread = d.
// Second streaming pass over x (L2-resident after kernel B).
// ---------------------------------------------------------------------------
__global__ void k_wsum(const float* __restrict__ x,
                       const float* __restrict__ Dcoef,
                       float* __restrict__ y) {
    __shared__ float dc[TT];
    int b = blockIdx.x;
    int d = threadIdx.x;
    for (int t = d; t < TT; t += 256) dc[t] = Dcoef[b * TT + t];
    __syncthreads();
    float acc = 0.f;
    #pragma unroll 4
    for (int t = 0; t < TT; ++t)
        acc += dc[t] * x[((size_t)t * BB + b) * DD + d];
    y[b * DD + d] = acc;
}

// ---------------------------------------------------------------------------
// Kernel E: out(64x256) = y(64x256) @ W(256x256) via V_WMMA_F32_16X16X4_F32.
// 64 wave-tiles of 16x16; K-loop of 64 steps (K=4 each).
// Fragment layouts per CDNA5 ISA 32-bit A/B/C tables:
//   A: lanes 0-15 hold (M=lane, K=k..k+1), lanes 16-31 hold (K=k+2..k+3)
//   B: lanes 0-15 hold (K=k..k+1, N=lane), lanes 16-31 hold (K=k+2..k+3)
//   C/D: VGPR v -> row m0 + 8*half + v, col n0 + (lane&15)
// ---------------------------------------------------------------------------
__global__ void k_gemm_wmma(const float* __restrict__ y,
                            const float* __restrict__ W,
                            float* __restrict__ out) {
    int wave = threadIdx.x >> 5;
    int lane = threadIdx.x & 31;
    int tile = blockIdx.x * 4 + wave;   // 0..63
    int mt = tile >> 4;                 // 0..3   (M tiles, M=64)
    int nt = tile & 15;                 // 0..15  (N tiles, N=256)
    int m0 = mt * 16, n0 = nt * 16;
    int half = lane >> 4;               // 0: K=k..k+1, 1: K=k+2..k+3
    int lr = lane & 15;

    v8f c = {};
    #pragma unroll 4
    for (int k = 0; k < DD; k += 4) {
        int ka = k + half * 2;
        v2f a, bf;
        a.x  = y[(m0 + lr) * DD + ka];
        a.y  = y[(m0 + lr) * DD + ka + 1];
        bf.x = W[(size_t)ka * FF + n0 + lr];
        bf.y = W[(size_t)(ka + 1) * FF + n0 + lr];
        c = __builtin_amdgcn_wmma_f32_16x16x4_f32(
                /*neg_a=*/false, a, /*neg_b=*/false, bf,
                /*c_mod=*/(short)0, c, /*reuse_a=*/false, /*reuse_b=*/false);
    }
    #pragma unroll
    for (int v = 0; v < 8; ++v) {
        int row = m0 + half * 8 + v;
        int col = n0 + lr;
        out[row * FF + col] = c[v];
    }
}

// ---------------------------------------------------------------------------
extern "C" void kernel_launch(void* const* d_in, const int* in_sizes, int n_in,
                              void* d_out, int out_size, void* d_ws, size_t ws_size,
                              hipStream_t stream) {
    (void)in_sizes; (void)n_in; (void)out_size; (void)ws_size;
    const float* x     = (const float*)d_in[0];   // (T,B,D) fp32
    const int*   turns = (const int*)d_in[1];     // (B,) int32
    const float* W     = (const float*)d_in[2];   // (D,F)
    const float* a1    = (const float*)d_in[3];   // (F,1)
    const float* a2    = (const float*)d_in[4];   // (F,1)
    float*       out   = (float*)d_out;           // (B,F)

    float* ws    = (float*)d_ws;
    float* wa    = ws;                            // 2*DD            = 512
    float* p1    = wa + 2 * DD;                   // B*T             = 32768
    float* p2    = p1 + BB * TT;                  // B*T             = 32768
    float* dcoef = p2 + BB * TT;                  // B*T             = 32768
    float* y     = dcoef + BB * TT;               // B*D             = 16384
    // total: 115200 floats = 450 KiB of workspace

    k_wa<<<1, 256, 0, stream>>>(W, a1, a2, wa);
    k_rowdots<<<(TT * BB) / 8, 256, 0, stream>>>(x, wa, p1, p2);
    k_attn<<<BB, 256, 0, stream>>>(p1, p2, turns, dcoef);
    k_wsum<<<BB, 256, 0, stream>>>(x, dcoef, y);
    k_gemm_wmma<<<16, 128, 0, stream>>>(y, W, out);
}